// PermutatedSoftSparse_37598143709723
// MI455X (gfx1250) — compile-verified
//
#include <hip/hip_runtime.h>

// PermutatedSoftSparse: out = RowPermCombine( ColPermCombine( weight * softmask(mask) ) )
// I = J = 4096, NP0 = NP1 = 10, fp32 throughout (memory-bound: ~320MB moved, ~0.77 GFLOP).
// Strategy: stage gather targets in LDS via the CDNA5 Tensor Data Mover so every element of
// the 10-way-fanout operands crosses HBM/L2 exactly once; gathers are served by LDS.

#define NI 4096
#define NJ 4096
#define NP 10

typedef __attribute__((ext_vector_type(4))) unsigned int v4u;
typedef __attribute__((ext_vector_type(8))) int          v8i;
typedef __attribute__((ext_vector_type(4))) int          v4i;

// Build a 2-D TDM descriptor (D#) and issue TENSOR_LOAD_TO_LDS.
// Element size fixed at 4 bytes (fp32). Dimensions/strides in elements.
// D# layout per CDNA5 ISA ch.8:
//   group0: [1:0]=count, [63:32]=lds_addr, [120:64]=global_addr, [127:126]=type(2)
//   group1: [17:16]=data_size(2 -> 4B), [79:48]=tensor_dim0, [111:80]=tensor_dim1,
//           [127:112]=tile_dim0, [143:128]=tile_dim1, [207:160]=tensor_dim0_stride
// This toolchain exposes the 6-arg builtin:
//   (uint32x4 g0, int32x8 g1, int32x4 g2, int32x4 g3, int32x8 extra, i32 cpol)
__device__ __forceinline__ void tdm_load_2d_f32(unsigned lds_off,
                                                const void* global_tile,
                                                unsigned tile_d0, unsigned tile_d1,
                                                unsigned tensor_d0, unsigned tensor_d1,
                                                unsigned stride0_elems) {
  unsigned long long ga = (unsigned long long)global_tile;
  v4u g0;
  g0.x = 1u;                                                   // count=1, user descriptor
  g0.y = lds_off;                                              // LDS byte address (wave-relative)
  g0.z = (unsigned)(ga & 0xffffffffull);                       // global_addr[31:0]
  g0.w = (unsigned)((ga >> 32) & 0x01ffffffull) | (2u << 30);  // global_addr[56:32] | type=2
  v8i g1;
  g1[0] = (int)(2u << 16);                                     // wg_mask=0, data_size=4B
  g1[1] = (int)((tensor_d0 & 0xffffu) << 16);                  // abar_addr=0 | tensor_dim0 lo16
  g1[2] = (int)((tensor_d0 >> 16) | ((tensor_d1 & 0xffffu) << 16));
  g1[3] = (int)((tensor_d1 >> 16) | ((tile_d0 & 0xffffu) << 16));
  g1[4] = (int)(tile_d1 & 0xffffu);                            // tile_dim1 | tile_dim2=0
  g1[5] = (int)stride0_elems;                                  // tensor_dim0_stride lo32
  g1[6] = 0;                                                   // stride0 hi16 | stride1 lo16
  g1[7] = 0;
  v4i g2; g2[0] = 0; g2[1] = 0; g2[2] = 0; g2[3] = 0;          // 2-D tensor: groups 2/3 unused
  v4i g3; g3[0] = 0; g3[1] = 0; g3[2] = 0; g3[3] = 0;
  v8i gx; gx[0] = 0; gx[1] = 0; gx[2] = 0; gx[3] = 0;          // trailing group (unused here)
  gx[4] = 0; gx[5] = 0; gx[6] = 0; gx[7] = 0;
  __builtin_amdgcn_tensor_load_to_lds(g0, g1, g2, g3, gx, 0);
}

// clip((XI - GAMMA) * sigmoid(x) + GAMMA, 0, 1) with XI=1.1, GAMMA=-0.1
__device__ __forceinline__ float soft_mask_f(float x) {
  float sig = 1.0f / (1.0f + __expf(-x));     // v_exp_f32 + v_rcp_f32
  float m = fmaf(1.2f, sig, -0.1f);
  return fminf(fmaxf(m, 0.0f), 1.0f);
}

// ---------------- Pass A: w1[i,j] = sum_p sw[i, perm1[p,j]] * s1[p,j] ----------------
// One workgroup per row i. TDM stages weight-row + mask-row (16KB each) into LDS,
// sw is computed in place, then the 10 column-gathers per output come from LDS.
__global__ void __launch_bounds__(256)
psparse_colperm_kernel(const float* __restrict__ weight,
                       const float* __restrict__ smask,
                       const float* __restrict__ scales1,
                       const int*   __restrict__ perm1,
                       float*       __restrict__ w1) {
  extern __shared__ float lds_a[];     // [0:NJ) weight row -> sw row, [NJ:2NJ) mask row
  const int row = (int)blockIdx.x;
  const unsigned tid = threadIdx.x;

  if (tid < 32u) {                     // wave 0 drives the DMA
    unsigned base = (unsigned)(unsigned long long)lds_a;   // low 32 bits = LDS offset
    tdm_load_2d_f32(base,             weight + (size_t)row * NJ, NJ, 1, NJ, 1, NJ);
    tdm_load_2d_f32(base + NJ * 4u,   smask  + (size_t)row * NJ, NJ, 1, NJ, 1, NJ);
    __builtin_amdgcn_s_wait_tensorcnt(0);
  }
  __syncthreads();

  // sw = weight * softmask(mask), in place (coalesced LDS, one sigmoid per element)
  #pragma unroll
  for (int t = 0; t < NJ / 256; ++t) {
    int k = (int)tid + 256 * t;
    lds_a[k] = lds_a[k] * soft_mask_f(lds_a[NJ + k]);
  }
  __syncthreads();

  // 10-way column gather-combine; perm1/scales1 loads are fully coalesced over j
  for (int t = 0; t < NJ / 256; ++t) {
    int j = (int)tid + 256 * t;
    float acc = 0.0f;
    #pragma unroll
    for (int p = 0; p < NP; ++p) {
      int c = perm1[p * NJ + j];
      acc = fmaf(lds_a[c], scales1[p * NJ + j], acc);
    }
    w1[(size_t)row * NJ + j] = acc;
  }
}

// ---------------- Pass B: out[i,j] = sum_p w1[perm0[p,i], j] * s0[p,i] ----------------
// One workgroup per 16-column chunk. TDM stages the full 4096x16 column slab of w1
// (256KB — fits CDNA5's 320KB LDS) so each w1 element is read from L2/HBM once and
// the 10x row-gather fan-out is served entirely from LDS.
__global__ void __launch_bounds__(256)
psparse_rowperm_kernel(const float* __restrict__ w1,
                       const float* __restrict__ scales0,
                       const int*   __restrict__ perm0,
                       float*       __restrict__ out) {
  extern __shared__ float lds_b[];     // NI rows x 16 cols, row-major
  const int j0 = (int)blockIdx.x * 16;
  const unsigned tid = threadIdx.x;

  if (tid < 32u) {
    unsigned base = (unsigned)(unsigned long long)lds_b;
    // tile: 16 contiguous floats (64B) per row, 4096 rows, row stride NJ elements
    tdm_load_2d_f32(base, w1 + j0, 16, NI, NJ, NI, NJ);
    __builtin_amdgcn_s_wait_tensorcnt(0);
  }
  __syncthreads();

  const int c  = (int)(tid & 15u);     // column within chunk
  const int i0 = (int)(tid >> 4u);     // 16 row phases per iteration
  for (int t = 0; t < NI / 16; ++t) {
    int i = i0 + 16 * t;
    float acc = 0.0f;
    #pragma unroll
    for (int p = 0; p < NP; ++p) {
      int r = perm0[p * NI + i];                       // broadcast across the 16 lanes sharing i
      acc = fmaf(lds_b[r * 16 + c], scales0[p * NI + i], acc);
    }
    out[(size_t)i * NJ + (size_t)(j0 + c)] = acc;      // 64B contiguous per half-wave
  }
}

extern "C" void kernel_launch(void* const* d_in, const int* in_sizes, int n_in,
                              void* d_out, int out_size, void* d_ws, size_t ws_size,
                              hipStream_t stream) {
  const float* weight = (const float*)d_in[0];   // (I, J) fp32
  const float* smask  = (const float*)d_in[1];   // (I, J) fp32
  const float* s0     = (const float*)d_in[2];   // (NP0, I) fp32
  const float* s1     = (const float*)d_in[3];   // (NP1, J) fp32
  const int*   p0     = (const int*)d_in[4];     // (NP0, I) int32
  const int*   p1     = (const int*)d_in[5];     // (NP1, J) int32
  float*       outp   = (float*)d_out;           // (I, J) fp32
  float*       w1     = (float*)d_ws;            // 64MB intermediate in workspace

  (void)in_sizes; (void)n_in; (void)out_size; (void)ws_size;

  // Pass A: one block per row; 32KB dynamic LDS (weight row + mask row)
  psparse_colperm_kernel<<<NI, 256, 2u * NJ * sizeof(float), stream>>>(weight, smask, s1, p1, w1);
  // Pass B: one block per 16-column chunk; 256KB dynamic LDS (full column slab of w1)
  psparse_rowperm_kernel<<<NJ / 16, 256, (size_t)NI * 16u * sizeof(float), stream>>>(w1, s0, p0, outp);
}